// CrossCorrelation_80358838108820
// MI455X (gfx1250) — compile-verified
//
#include <hip/hip_runtime.h>
#include <math.h>

// CrossCorrelation: B=128, T=1024, D=512, f32.
// Memory-bound kernel (~768MB HBM traffic, <1 GFLOP). One workgroup per batch:
// k[b] (2MB) is re-read in the WMMA dot phase from L2. wt dot products use
// V_WMMA_F32_16X16X4_F32 (full f32 precision matrix path).

typedef float v2f __attribute__((ext_vector_type(2)));
typedef float v8f __attribute__((ext_vector_type(8)));

#define TN 1024
#define DN 512
#define EPSC 1e-5f

__global__ __launch_bounds__(1024)
void cc_kernel(const float* __restrict__ k, const float* __restrict__ q,
               const float* __restrict__ v, float* __restrict__ out)
{
    __shared__ __align__(16) float redA[1024];
    __shared__ __align__(16) float redB[1024];
    __shared__ __align__(16) float c_lds[DN];    // qn_last / (sigma_k + eps)
    __shared__ __align__(16) float mk_lds[DN];   // mean of k over time
    __shared__ __align__(16) float sk_lds[DN];   // std of k over time
    __shared__ __align__(16) float wt_lds[TN];
    __shared__ float constb_s;
    __shared__ float winv_s;

    const int tid  = threadIdx.x;
    const int b    = blockIdx.x;
    const int d    = tid & (DN - 1);
    const int half = tid >> 9;                   // 0 or 1 (two t-halves)
    const size_t base = (size_t)b * TN * DN;

    // ---------------- Phase A1: per-d stats of k over time ----------------
    {
        const float* kp = k + base + (size_t)(half * 512) * DN + d;
        float s = 0.f, ss = 0.f;
        #pragma unroll 8
        for (int i = 0; i < 512; ++i) {
            float x = kp[(size_t)i * DN];
            s += x; ss += x * x;
        }
        redA[tid] = s; redB[tid] = ss;
    }
    __syncthreads();
    if (half == 0) {
        float st  = redA[tid] + redA[tid + 512];
        float sst = redB[tid] + redB[tid + 512];
        float m   = st * (1.f / TN);
        float var = sst * (1.f / TN) - m * m;
        mk_lds[d] = m;
        sk_lds[d] = sqrtf(fmaxf(var, 0.f));
    }
    __syncthreads();

    // ---------------- Phase A2: per-d stats of q, build c[d] ----------------
    {
        const float* qp = q + base + (size_t)(half * 512) * DN + d;
        float s = 0.f, ss = 0.f;
        #pragma unroll 8
        for (int i = 0; i < 512; ++i) {
            float x = qp[(size_t)i * DN];
            s += x; ss += x * x;
        }
        redA[tid] = s; redB[tid] = ss;
    }
    __syncthreads();
    if (half == 0) {
        float st  = redA[tid] + redA[tid + 512];
        float sst = redB[tid] + redB[tid + 512];
        float m   = st * (1.f / TN);
        float var = sst * (1.f / TN) - m * m;
        float sq_ = sqrtf(fmaxf(var, 0.f));
        float qn  = (q[base + (size_t)(TN - 1) * DN + d] - m) / (sq_ + EPSC);
        c_lds[d]  = qn / (sk_lds[d] + EPSC);
    }
    __syncthreads();

    // ---------------- constb = sum_d mk[d]*c[d] ----------------
    redA[tid] = (tid < DN) ? mk_lds[tid] * c_lds[tid] : 0.f;
    __syncthreads();
    for (int off = 512; off > 0; off >>= 1) {
        if (tid < off) redA[tid] += redA[tid + off];
        __syncthreads();
    }
    if (tid == 0) constb_s = redA[0];
    __syncthreads();
    const float constb = constb_s;

    // ---------------- Phase B: wt[t] = exp((dot(k[t],c) - constb)/T) -------
    // V_WMMA_F32_16X16X4_F32: A = 16(t) x 4(d) tile of k, B = c chunk
    // replicated into all 16 columns. Accumulate over d in steps of 4.
    // A layout: lane<16 -> M=lane, K=0,1 ; lane>=16 -> M=lane-16, K=2,3.
    // B layout: lane<16 -> N=lane, K=0,1 ; lane>=16 -> N=lane-16, K=2,3.
    // C layout: vgpr r, lanes 0-15 -> M=r ; lanes 16-31 -> M=8+r.
    {
        const int lane = tid & 31;
        const int wave = tid >> 5;
        const int lrow = lane & 15;   // t offset within tile
        const int lkhi = lane >> 4;   // selects K pair {0,1} vs {2,3}
        for (int tile = wave; tile < TN / 16; tile += 32) {
            const int t0 = tile * 16;
            const float* kfrag = k + base + (size_t)(t0 + lrow) * DN + 2 * lkhi;
            v8f acc = {0.f, 0.f, 0.f, 0.f, 0.f, 0.f, 0.f, 0.f};
            #pragma unroll 4
            for (int d0 = 0; d0 < DN; d0 += 4) {
                v2f a  = *(const v2f*)(kfrag + d0);                 // L2-resident
                v2f bf = *(const v2f*)(&c_lds[d0 + 2 * lkhi]);      // LDS broadcast
                acc = __builtin_amdgcn_wmma_f32_16x16x4_f32(
                        /*neg_a=*/false, a, /*neg_b=*/false, bf,
                        /*c_mod=*/(short)0, acc,
                        /*reuse_a=*/false, /*reuse_b=*/false);
            }
            if (lrow == 0) {  // lane 0 holds M=0..7, lane 16 holds M=8..15 (N=0)
                const int mbase = t0 + 8 * lkhi;
                #pragma unroll
                for (int r = 0; r < 8; ++r)
                    wt_lds[mbase + r] = expf((acc[r] - constb) * (1.f / TN));
            }
        }
    }
    __syncthreads();

    // ---------------- wsum over t = 0..T-2 ----------------
    redA[tid] = (tid < TN - 1) ? wt_lds[tid] : 0.f;
    __syncthreads();
    for (int off = 512; off > 0; off >>= 1) {
        if (tid < off) redA[tid] += redA[tid + off];
        __syncthreads();
    }
    if (tid == 0) winv_s = 1.f / redA[0];
    __syncthreads();
    const float winv = winv_s;

    // ---------------- Phase C: out[d] = winv * sum_t wt[t]*v[b,t+1,d] ------
    {
        const int tstart = half * 512;
        const int tend   = half ? (TN - 1) : 512;    // half0: 0..511, half1: 512..1022
        const float* vp  = v + base + (size_t)(tstart + 1) * DN + d;
        float acc = 0.f;
        #pragma unroll 4
        for (int t = tstart; t < tend; ++t)
            acc += wt_lds[t] * vp[(size_t)(t - tstart) * DN];
        redB[tid] = acc;
    }
    __syncthreads();
    if (half == 0)
        out[(size_t)b * DN + d] = (redB[tid] + redB[tid + 512]) * winv;
}

extern "C" void kernel_launch(void* const* d_in, const int* in_sizes, int n_in,
                              void* d_out, int out_size, void* d_ws, size_t ws_size,
                              hipStream_t stream) {
    const float* k = (const float*)d_in[0];
    const float* q = (const float*)d_in[1];
    const float* v = (const float*)d_in[2];
    float* out = (float*)d_out;
    (void)in_sizes; (void)n_in; (void)out_size; (void)d_ws; (void)ws_size;
    cc_kernel<<<dim3(128), dim3(1024), 0, stream>>>(k, q, v, out);
}